// ResPointBlock_3504693313574
// MI455X (gfx1250) — compile-verified
//
#include <hip/hip_runtime.h>
#include <hip/hip_bf16.h>
#include <math.h>

typedef __attribute__((ext_vector_type(2))) float v2f;
typedef __attribute__((ext_vector_type(8))) float v8f;

#define CIN 61
#define COUT 64
#define EPS 1e-5f
#define ROWP 66   // 64 + 2 pad floats: keeps ds_load_b64 A-frag fetches conflict-free

// Native fp32 atomic max (no CAS loop): GLOBAL_ATOMIC_MAX_NUM_F32, no return (STOREcnt).
__device__ __forceinline__ void atomic_max_f32(float* addr, float val) {
  asm volatile("global_atomic_max_num_f32 %0, %1, off" :: "v"(addr), "v"(val) : "memory");
}

__device__ __forceinline__ v8f wmma4(v2f a, v2f b, v8f c) {
  // V_WMMA_F32_16X16X4_F32: D = A(16x4) * B(4x16) + C(16x16)
  return __builtin_amdgcn_wmma_f32_16x16x4_f32(false, a, false, b, (short)0, c,
                                               false, false);
}

// ---------------------------------------------------------------------------
// Init: d_out = -inf (segment_max identity), zero BN stat accumulators.
// ---------------------------------------------------------------------------
__global__ void init_kernel(float* __restrict__ out, float* __restrict__ stats,
                            int total) {
  int g = blockIdx.x * blockDim.x + threadIdx.x;
  int stride = gridDim.x * blockDim.x;
  for (int i = g; i < total; i += stride) out[i] = -INFINITY;
  if (g < 1024) stats[g] = 0.0f;
}

// ---------------------------------------------------------------------------
// Shortcut GEMM: ys = x @ Ws + bs  (N x 64, K = 61 zero-padded to 64),
// accumulating per-channel sum / sumsq for its BatchNorm.
// ---------------------------------------------------------------------------
__global__ __launch_bounds__(128) void node_gemm_kernel(
    const float* __restrict__ x, const float* __restrict__ Wsm,
    const float* __restrict__ bsv, float* __restrict__ ys,
    float* __restrict__ stats, int N) {
  __shared__ __align__(16) float tileS[16 * ROWP];

  const int lane = threadIdx.x & 31;
  const int wv = threadIdx.x >> 5;
  const int n0 = wv * 16;
  const int nn = n0 + (lane & 15);
  const int kh = (lane >> 4) * 2;       // K sub-offset per lane half
  const int mh = (lane >> 4) * 8;       // C/D row offset per lane half

  // Preload B fragments (Ws is 61x64; rows 61..63 are zero).
  v2f Bf[16];
#pragma unroll
  for (int k = 0; k < 16; ++k) {
    int r0 = 4 * k + kh, r1 = r0 + 1;
    Bf[k].x = (r0 < CIN) ? Wsm[r0 * COUT + nn] : 0.0f;
    Bf[k].y = (r1 < CIN) ? Wsm[r1 * COUT + nn] : 0.0f;
  }
  const float bias = bsv[nn];

  float sAcc = 0.0f, qAcc = 0.0f;
  const int numTiles = (N + 15) >> 4;
  const int r = threadIdx.x >> 3;
  const int c0 = (threadIdx.x & 7) * 8;

  for (int tile = blockIdx.x; tile < numTiles; tile += gridDim.x) {
    const int e0 = tile << 4;
    // Cooperative load of 16 node rows (K padded with zeros).
    {
      const int node = e0 + r;
      const bool valid = node < N;
#pragma unroll
      for (int c = c0; c < c0 + 8; ++c) {
        float v = 0.0f;
        if (valid && c < CIN) v = x[(long long)node * CIN + c];
        tileS[r * ROWP + c] = v;
      }
    }
    __syncthreads();

    v8f acc = {};
#pragma unroll
    for (int k = 0; k < 16; ++k) {
      v2f a = *(const v2f*)&tileS[(lane & 15) * ROWP + 4 * k + kh];
      acc = wmma4(a, Bf[k], acc);
    }

#pragma unroll
    for (int j = 0; j < 8; ++j) {
      const int m = j + mh;
      if (e0 + m < N) {
        float y = acc[j] + bias;
        ys[(long long)(e0 + m) * COUT + nn] = y;
        sAcc += y;
        qAcc += y * y;
      }
    }
    __syncthreads();
  }

  sAcc += __shfl_xor(sAcc, 16, 32);
  qAcc += __shfl_xor(qAcc, 16, 32);
  if (lane < 16) {
    atomicAdd(&stats[nn], sAcc);
    atomicAdd(&stats[COUT + nn], qAcc);
  }
}

// ---------------------------------------------------------------------------
// Edge MLP passes (recompute strategy; x lives in L2 so gathers are cheap):
//  MODE 1: a1 = ReLU(msg@W1+b1)                      -> stats for BN1
//  MODE 2: h1 = BN1(a1); a2 = ReLU(h1@W2+b2)         -> stats for BN2
//  MODE 3: h2 = BN2(a2); atomic scatter-max to d_out
// ---------------------------------------------------------------------------
template <int MODE>
__global__ __launch_bounds__(128) void edge_pass_kernel(
    const float* __restrict__ x, const float* __restrict__ pos,
    const long long* __restrict__ src, const long long* __restrict__ dst,
    const float* __restrict__ W1, const float* __restrict__ b1,
    const float* __restrict__ W2, const float* __restrict__ b2,
    const float* __restrict__ bn1, const float* __restrict__ bn2,
    float* __restrict__ stats, float* __restrict__ out, int E) {
  __shared__ __align__(16) float msgS[16 * ROWP];
  __shared__ __align__(16) float h1S[16 * ROWP];
  __shared__ long long dstS[16];

  const int lane = threadIdx.x & 31;
  const int wv = threadIdx.x >> 5;
  const int n0 = wv * 16;
  const int nn = n0 + (lane & 15);
  const int kh = (lane >> 4) * 2;
  const int mh = (lane >> 4) * 8;

  // Preload W fragments into registers; reused across the whole tile loop.
  v2f B1f[16], B2f[16];
#pragma unroll
  for (int k = 0; k < 16; ++k) {
    B1f[k].x = W1[(4 * k + kh) * COUT + nn];
    B1f[k].y = W1[(4 * k + kh + 1) * COUT + nn];
  }
  if (MODE >= 2) {
#pragma unroll
    for (int k = 0; k < 16; ++k) {
      B2f[k].x = W2[(4 * k + kh) * COUT + nn];
      B2f[k].y = W2[(4 * k + kh + 1) * COUT + nn];
    }
  }
  const float bias1 = b1[nn];
  const float bias2 = (MODE >= 2) ? b2[nn] : 0.0f;
  const float sc1 = (MODE >= 2) ? bn1[nn] : 0.0f;
  const float sh1 = (MODE >= 2) ? bn1[COUT + nn] : 0.0f;
  const float sc2 = (MODE == 3) ? bn2[nn] : 0.0f;
  const float sh2 = (MODE == 3) ? bn2[COUT + nn] : 0.0f;

  float sAcc = 0.0f, qAcc = 0.0f;
  const int numTiles = (E + 15) >> 4;
  const int r = threadIdx.x >> 3;
  const int c0 = (threadIdx.x & 7) * 8;

  for (int tile = blockIdx.x; tile < numTiles; tile += gridDim.x) {
    const int e0 = tile << 4;
    // Build the 16x64 message tile: [x[src] (61) | pos[src]-pos[dst] (3)].
    {
      const int e = e0 + r;
      const bool valid = e < E;
      const long long s_ = valid ? src[e] : 0;
      const long long d_ = valid ? dst[e] : 0;
#pragma unroll
      for (int c = c0; c < c0 + 8; ++c) {
        float v = 0.0f;
        if (valid) {
          if (c < CIN)
            v = x[s_ * CIN + c];
          else
            v = pos[s_ * 3 + (c - CIN)] - pos[d_ * 3 + (c - CIN)];
        }
        msgS[r * ROWP + c] = v;
      }
      if (MODE == 3 && (threadIdx.x & 7) == 0) dstS[r] = d_;
    }
    __syncthreads();

    // GEMM1: 16 chained k=4 WMMAs.
    v8f c1 = {};
#pragma unroll
    for (int k = 0; k < 16; ++k) {
      v2f a = *(const v2f*)&msgS[(lane & 15) * ROWP + 4 * k + kh];
      c1 = wmma4(a, B1f[k], c1);
    }
#pragma unroll
    for (int j = 0; j < 8; ++j) c1[j] = fmaxf(c1[j] + bias1, 0.0f);

    if (MODE == 1) {
#pragma unroll
      for (int j = 0; j < 8; ++j) {
        if (e0 + j + mh < E) {
          sAcc += c1[j];
          qAcc += c1[j] * c1[j];
        }
      }
    } else {
      // BN1 affine, exchange h1 through LDS (wave w holds N-slice == K-slice
      // needed by every wave for GEMM2).
#pragma unroll
      for (int j = 0; j < 8; ++j) {
        float h = sc1 * c1[j] + sh1;
        h1S[(j + mh) * ROWP + nn] = h;
      }
      __syncthreads();

      v8f c2 = {};
#pragma unroll
      for (int k = 0; k < 16; ++k) {
        v2f a = *(const v2f*)&h1S[(lane & 15) * ROWP + 4 * k + kh];
        c2 = wmma4(a, B2f[k], c2);
      }
#pragma unroll
      for (int j = 0; j < 8; ++j) c2[j] = fmaxf(c2[j] + bias2, 0.0f);

      if (MODE == 2) {
#pragma unroll
        for (int j = 0; j < 8; ++j) {
          if (e0 + j + mh < E) {
            sAcc += c2[j];
            qAcc += c2[j] * c2[j];
          }
        }
      } else {  // MODE 3: BN2 + native float scatter-max
#pragma unroll
        for (int j = 0; j < 8; ++j) {
          const int m = j + mh;
          if (e0 + m < E) {
            float h2 = sc2 * c2[j] + sh2;
            atomic_max_f32(&out[dstS[m] * COUT + nn], h2);
          }
        }
      }
    }
    __syncthreads();
  }

  if (MODE != 3) {
    sAcc += __shfl_xor(sAcc, 16, 32);
    qAcc += __shfl_xor(qAcc, 16, 32);
    if (lane < 16) {
      atomicAdd(&stats[nn], sAcc);
      atomicAdd(&stats[COUT + nn], qAcc);
    }
  }
}

// ---------------------------------------------------------------------------
// Fold (sum, sumsq) -> per-channel (scale, shift): BN(v) = scale*v + shift.
// ---------------------------------------------------------------------------
__global__ void finalize_bn_kernel(const float* __restrict__ stats,
                                   const float* __restrict__ g,
                                   const float* __restrict__ be,
                                   float* __restrict__ scsh, float count) {
  int t = threadIdx.x;
  if (t < COUT) {
    float m = stats[t] / count;
    float v = stats[COUT + t] / count - m * m;
    float sc = g[t] * rsqrtf(v + EPS);
    scsh[t] = sc;
    scsh[COUT + t] = be[t] - m * sc;
  }
}

// ---------------------------------------------------------------------------
// agg: -inf -> 0 (nodes with no in-edges) + stats for the final BN.
// Requires gridDim.x*blockDim.x % 64 == 0 so each thread's channel is fixed.
// ---------------------------------------------------------------------------
__global__ __launch_bounds__(256) void fix_agg_kernel(float* __restrict__ out,
                                                      float* __restrict__ stats,
                                                      int total) {
  __shared__ float ssh[256], qsh[256];
  const int g = blockIdx.x * blockDim.x + threadIdx.x;
  const int stride = gridDim.x * blockDim.x;
  float s = 0.0f, q = 0.0f;
  for (int i = g; i < total; i += stride) {
    float v = out[i];
    if (v == -INFINITY) v = 0.0f;
    out[i] = v;
    s += v;
    q += v * v;
  }
  ssh[threadIdx.x] = s;
  qsh[threadIdx.x] = q;
  __syncthreads();
  if (threadIdx.x < 128) {
    ssh[threadIdx.x] += ssh[threadIdx.x + 128];
    qsh[threadIdx.x] += qsh[threadIdx.x + 128];
  }
  __syncthreads();
  if (threadIdx.x < 64) {
    float fs = ssh[threadIdx.x] + ssh[threadIdx.x + 64];
    float fq = qsh[threadIdx.x] + qsh[threadIdx.x + 64];
    atomicAdd(&stats[threadIdx.x], fs);          // channel == threadIdx.x
    atomicAdd(&stats[COUT + threadIdx.x], fq);
  }
}

// ---------------------------------------------------------------------------
// out = ReLU(BNf(agg) + BNs(ys))
// ---------------------------------------------------------------------------
__global__ void final_combine_kernel(float* __restrict__ out,
                                     const float* __restrict__ ys,
                                     const float* __restrict__ bnF,
                                     const float* __restrict__ bnS, int total) {
  const int g = blockIdx.x * blockDim.x + threadIdx.x;
  const int stride = gridDim.x * blockDim.x;
  for (int i = g; i < total; i += stride) {
    const int c = i & (COUT - 1);
    float v = bnF[c] * out[i] + bnF[COUT + c] + bnS[c] * ys[i] + bnS[COUT + c];
    out[i] = fmaxf(v, 0.0f);
  }
}

// ---------------------------------------------------------------------------
extern "C" void kernel_launch(void* const* d_in, const int* in_sizes, int n_in,
                              void* d_out, int out_size, void* d_ws,
                              size_t ws_size, hipStream_t stream) {
  const float* x = (const float*)d_in[0];
  const float* pos = (const float*)d_in[1];
  const long long* ei = (const long long*)d_in[2];
  const float* W1 = (const float*)d_in[3];
  const float* b1 = (const float*)d_in[4];
  const float* g1 = (const float*)d_in[5];
  const float* be1 = (const float*)d_in[6];
  const float* W2 = (const float*)d_in[7];
  const float* b2 = (const float*)d_in[8];
  const float* g2 = (const float*)d_in[9];
  const float* be2 = (const float*)d_in[10];
  const float* Wsm = (const float*)d_in[11];
  const float* bsv = (const float*)d_in[12];
  const float* gs = (const float*)d_in[13];
  const float* bes = (const float*)d_in[14];
  const float* gf = (const float*)d_in[15];
  const float* bef = (const float*)d_in[16];

  const int N = in_sizes[0] / CIN;
  const int E = in_sizes[2] / 2;
  const long long* src = ei;
  const long long* dst = ei + E;
  float* out = (float*)d_out;

  float* ws = (float*)d_ws;
  float* st1 = ws;           // [128] sum/sumsq BN1
  float* st2 = ws + 128;     // [128] BN2
  float* stS = ws + 256;     // [128] shortcut BN
  float* stF = ws + 384;     // [128] final BN
  float* bn1 = ws + 512;     // [128] scale/shift
  float* bn2 = ws + 640;
  float* bnS = ws + 768;
  float* bnF = ws + 896;
  float* ys = ws + 1024;     // [N*64] shortcut pre-BN activations

  const int total = N * COUT;
  const int edgeTiles = (E + 15) >> 4;
  const int nodeTiles = (N + 15) >> 4;
  const int edgeBlocks = edgeTiles < 2048 ? edgeTiles : 2048;
  const int nodeBlocks = nodeTiles < 1024 ? nodeTiles : 1024;

  init_kernel<<<2048, 256, 0, stream>>>(out, ws, total);

  node_gemm_kernel<<<nodeBlocks, 128, 0, stream>>>(x, Wsm, bsv, ys, stS, N);
  finalize_bn_kernel<<<1, 64, 0, stream>>>(stS, gs, bes, bnS, (float)N);

  edge_pass_kernel<1><<<edgeBlocks, 128, 0, stream>>>(
      x, pos, src, dst, W1, b1, W2, b2, bn1, bn2, st1, out, E);
  finalize_bn_kernel<<<1, 64, 0, stream>>>(st1, g1, be1, bn1, (float)E);

  edge_pass_kernel<2><<<edgeBlocks, 128, 0, stream>>>(
      x, pos, src, dst, W1, b1, W2, b2, bn1, bn2, st2, out, E);
  finalize_bn_kernel<<<1, 64, 0, stream>>>(st2, g2, be2, bn2, (float)E);

  edge_pass_kernel<3><<<edgeBlocks, 128, 0, stream>>>(
      x, pos, src, dst, W1, b1, W2, b2, bn1, bn2, st2, out, E);

  fix_agg_kernel<<<512, 256, 0, stream>>>(out, stF, total);
  finalize_bn_kernel<<<1, 64, 0, stream>>>(stF, gf, bef, bnF, (float)N);

  final_combine_kernel<<<2048, 256, 0, stream>>>(out, ys, bnF, bnS, total);
}